// gcc_Conv2d_64347200028713
// MI455X (gfx1250) — compile-verified
//
#include <hip/hip_runtime.h>

typedef __attribute__((ext_vector_type(2))) float v2f;
typedef __attribute__((ext_vector_type(8))) float v8f;

namespace {
constexpr int kC     = 384;
constexpr int kH     = 56;
constexpr int kW     = 56;
constexpr int kB     = 32;
constexpr int kLda   = 60;   // padded LDS row stride (floats): conflict-free column reads
constexpr int kBPerWG = 8;   // batches handled per workgroup (amortizes circulant build)
}

__global__ __launch_bounds__(128, 4) void circ_dwconv_wmma_f32(
    const float* __restrict__ x,
    const float* __restrict__ wt,
    const float* __restrict__ bias,
    float* __restrict__ out)
{
    __shared__ float s_w[kH];
    __shared__ float s_A[64 * kLda];   // circulant, M padded to 64 with zero rows

    const int c    = blockIdx.x;
    const int tid  = threadIdx.x;
    const int lane = tid & 31;
    const int wv   = tid >> 5;         // wave id == N-tile id (0..3)

    // Stage per-channel weights, then build circulant A[m][k] = w[(k-m) mod 56]
    if (tid < kH) s_w[tid] = wt[c * kH + tid];
    __syncthreads();

    for (int i = tid; i < 64 * kH; i += 128) {
        const int m = i / kH;
        const int k = i - m * kH;
        float v = 0.0f;
        if (m < kH) {
            int d = k - m;
            if (d < 0) d += kH;
            v = s_w[d];
        }
        s_A[m * kLda + k] = v;
    }
    __syncthreads();

    const float bv  = bias[c];
    const int nloc  = lane & 15;       // N within tile / M within tile (A side)
    const int half  = lane >> 4;       // hi/lo half of wave
    const int n     = wv * 16 + nloc;  // output column 0..63
    const bool nok  = (n < kW);

    // lane-invariant pieces of addressing, hoisted
    const int bcol  = n;               // x column this lane streams

    #pragma unroll 1
    for (int bi = 0; bi < kBPerWG; ++bi) {
        const int b = blockIdx.y * kBPerWG + bi;
        const size_t base = (((size_t)b * kC + c) * kH) * kW;
        const float* __restrict__ xp = x + base;
        float* __restrict__ op = out + base;

        v8f acc[4] = {};

        // K = 56 = 14 exact steps of 4; full unroll -> deep load pipelining
        #pragma unroll
        for (int ks = 0; ks < 14; ++ks) {
            const int k0 = ks * 4 + half * 2;    // K row group for this half-wave
            // B fragment (4x16 fp32): rows striped across lanes, streamed from HBM
            v2f bf;
            bf.x = nok ? __builtin_nontemporal_load(xp + k0 * kW + bcol) : 0.0f;
            bf.y = nok ? __builtin_nontemporal_load(xp + (k0 + 1) * kW + bcol) : 0.0f;

            #pragma unroll
            for (int mt = 0; mt < 4; ++mt) {
                // A fragment (16x4 fp32): lane holds row m = mt*16 + nloc
                const float* ap = &s_A[(mt * 16 + nloc) * kLda + k0];
                v2f af;
                af.x = ap[0];
                af.y = ap[1];
                acc[mt] = __builtin_amdgcn_wmma_f32_16x16x4_f32(
                    false, af, false, bf, (short)0, acc[mt], false, false);
            }
        }

        // C/D layout: VGPR r, lanes<16 -> M = mt*16 + r, lanes>=16 -> M = mt*16 + 8 + r
        #pragma unroll
        for (int mt = 0; mt < 4; ++mt) {
            #pragma unroll
            for (int r = 0; r < 8; ++r) {
                const int m = mt * 16 + half * 8 + r;
                if (m < kH && nok)
                    __builtin_nontemporal_store(acc[mt][r] + bv, op + m * kW + n);
            }
        }
    }
}

extern "C" void kernel_launch(void* const* d_in, const int* in_sizes, int n_in,
                              void* d_out, int out_size, void* d_ws, size_t ws_size,
                              hipStream_t stream) {
    const float* x    = (const float*)d_in[0];  // [32,384,56,56] f32
    const float* wt   = (const float*)d_in[1];  // [384,1,56,1]   f32
    const float* bias = (const float*)d_in[2];  // [384]          f32
    float* out        = (float*)d_out;          // [32,384,56,56] f32

    dim3 grid(kC, kB / kBPerWG);   // (384, 4)
    circ_dwconv_wmma_f32<<<grid, 128, 0, stream>>>(x, wt, bias, out);
}